// RoIPooling_29686813950520
// MI455X (gfx1250) — compile-verified
//
#include <hip/hip_runtime.h>
#include <cstdint>

// RoI max pooling, exact reproduction of the JAX reference semantics.
//   features: [1, C, H, W] f32   (N == 1)
//   rois:     [R, 4] int32  (x1, y1, x2, y2 inclusive)
//   out:      [R, C, P, P] f32, P == 7
//
// MI455X strategy: pure data movement (feature map is L2-resident: 115 MB <
// 192 MB L2). One 64-thread block per (roi, 8-channel group). ROI geometry is
// computed once per block; the CDNA5 Tensor Data Mover streams each channel's
// ROI rectangle (<= 64x64 f32 = 16 KB) into one of two LDS buffers
// (tensor_load_to_lds, TENSORcnt), double-buffered so the DMA of channel k+1
// overlaps the 49-lane 7x7 bin reduction of channel k. Output is written as
// coalesced 49-float bursts.

#define P_BINS   7
#define H_FEAT   336
#define W_FEAT   336
#define TILE_MAX 64
#define CPB      8     // channels per block

typedef unsigned int v4u __attribute__((ext_vector_type(4)));
typedef int          v4i __attribute__((ext_vector_type(4)));
typedef int          v8i __attribute__((ext_vector_type(8)));

// b(i) = floor(lo + scale*i) with f32 mul and add rounded separately
// (matches jnp: lo + (size/P)*w, no FMA contraction).
__device__ __forceinline__ int bin_bound(float lo, float scale, int i) {
    float t = __fmul_rn(scale, (float)i);
    float v = __fadd_rn(lo, t);
    return (int)floorf(v);   // coords are non-negative: trunc == floor
}

// One TDM descriptor: 2D tile (tile_w x tile_h elems, 4B each) from global
// (row stride = stride_elems) into LDS at lds_byte_addr, tightly packed.
__device__ __forceinline__ void tdm_load_2d(unsigned lds_byte_addr,
                                            const void* gptr,
                                            int tile_w, int tile_h,
                                            int tensor_w, int tensor_h,
                                            unsigned long long stride_elems) {
    unsigned long long ga = (unsigned long long)(uintptr_t)gptr;
    v4u g0;
    g0[0] = 1u;                                       // count=1, user D#
    g0[1] = lds_byte_addr;                            // lds_addr [63:32]
    g0[2] = (unsigned)(ga & 0xFFFFFFFFu);             // global_addr [95:64]
    g0[3] = (unsigned)((ga >> 32) & 0x01FFFFFFu)      // global_addr [120:96]
          | 0x80000000u;                              // type=2 [127:126]
    v8i g1;
    g1[0] = 0x00020000;                               // data_size=2 (4 bytes)
    g1[1] = (tensor_w & 0xFFFF) << 16;                // tensor_dim0 [79:48]
    g1[2] = ((tensor_w >> 16) & 0xFFFF)
          | ((tensor_h & 0xFFFF) << 16);              // tensor_dim1 [111:80]
    g1[3] = ((tensor_h >> 16) & 0xFFFF)
          | ((tile_w & 0xFFFF) << 16);                // tile_dim0 [127:112]
    g1[4] = (tile_h & 0xFFFF);                        // tile_dim1; tile_dim2=0
    g1[5] = (int)(stride_elems & 0xFFFFFFFFull);      // tensor_dim0_stride lo
    g1[6] = (int)((stride_elems >> 32) & 0xFFFFull);  // stride hi; dim1_stride=0
    g1[7] = 0;
    v4i z4 = {0, 0, 0, 0};
#if defined(__clang_major__) && (__clang_major__ >= 23)
    v8i z8 = {0, 0, 0, 0, 0, 0, 0, 0};
    __builtin_amdgcn_tensor_load_to_lds(g0, g1, z4, z4, z8, 0);
#else
    __builtin_amdgcn_tensor_load_to_lds(g0, g1, z4, z4, 0);
#endif
}

__global__ __launch_bounds__(64)
void roi_maxpool_kernel(const float* __restrict__ feat,
                        const int*  __restrict__ rois,
                        float* __restrict__ out,
                        int C, int R) {
    const int c0 = blockIdx.x * CPB;
    const int r  = blockIdx.y;
    if (r >= R || c0 >= C) return;
    const int t = threadIdx.x;
    const int nch = (C - c0 < CPB) ? (C - c0) : CPB;

    __shared__ float tile[2][TILE_MAX * TILE_MAX];    // 2 x 16 KB, pitch=cols

    // ---- ROI geometry: once per block, reused for all channels ---------
    const int x1 = rois[r * 4 + 0];
    const int y1 = rois[r * 4 + 1];
    const int x2 = rois[r * 4 + 2];
    const int y2 = rois[r * 4 + 3];
    int rwi = x2 - x1 + 1; if (rwi < 1) rwi = 1;
    int rhi = y2 - y1 + 1; if (rhi < 1) rhi = 1;
    const float x1f = (float)x1, y1f = (float)y1;
    const float sx = __fdiv_rn((float)rwi, (float)P_BINS);
    const float sy = __fdiv_rn((float)rhi, (float)P_BINS);

    const int x0 = x1;                                // bin_bound(x1f,sx,0)==x1
    const int y0 = y1;
    int xend = bin_bound(x1f, sx, P_BINS); if (xend > W_FEAT) xend = W_FEAT;
    int yend = bin_bound(y1f, sy, P_BINS); if (yend > H_FEAT) yend = H_FEAT;
    int cols = xend - x0; cols = cols < 0 ? 0 : (cols > TILE_MAX ? TILE_MAX : cols);
    int rows = yend - y0; rows = rows < 0 ? 0 : (rows > TILE_MAX ? TILE_MAX : rows);

    // Per-lane bin window (channel-independent).
    const bool active = (t < P_BINS * P_BINS);
    int xlo = 0, xhi = 0, ylo = 0, yhi = 0, ph = 0, pw = 0;
    if (active) {
        pw = t % P_BINS;
        ph = t / P_BINS;
        xlo = bin_bound(x1f, sx, pw)     - x0;
        xhi = bin_bound(x1f, sx, pw + 1) - x0;
        ylo = bin_bound(y1f, sy, ph)     - y0;
        yhi = bin_bound(y1f, sy, ph + 1) - y0;
        if (xlo < 0) xlo = 0;
        if (ylo < 0) ylo = 0;
        if (xhi > cols) xhi = cols;
        if (yhi > rows) yhi = rows;
    }

    const size_t plane = (size_t)H_FEAT * (size_t)W_FEAT;
    const float* gbase = feat + (size_t)y0 * W_FEAT + x0;

    // ---- Double-buffered TDM pipeline over channels --------------------
    // TDM ignores EXEC and issues once per wave that reaches it, so only the
    // wave containing lane 0 may reach the tensor instruction.
    if (t == 0) {
        tdm_load_2d((unsigned)(uintptr_t)&tile[0][0],
                    gbase + (size_t)c0 * plane,
                    cols, rows, W_FEAT, H_FEAT, (unsigned long long)W_FEAT);
    }

    for (int k = 0; k < nch; ++k) {
        if (t == 0) {
            if (k + 1 < nch) {
                // Prefetch next channel into the other buffer, then wait for
                // the current one (TENSORcnt completes in-order per wave).
                tdm_load_2d((unsigned)(uintptr_t)&tile[(k + 1) & 1][0],
                            gbase + (size_t)(c0 + k + 1) * plane,
                            cols, rows, W_FEAT, H_FEAT,
                            (unsigned long long)W_FEAT);
                __builtin_amdgcn_s_wait_tensorcnt(1);
            } else {
                __builtin_amdgcn_s_wait_tensorcnt(0);
            }
        }
        __syncthreads();   // current buffer ready for all lanes

        if (active) {
            const float* buf = &tile[k & 1][0];
            float acc = -__builtin_huge_valf();       // segment_max identity
            for (int y = ylo; y < yhi; ++y) {
                const float* trow = buf + y * cols;
                for (int x = xlo; x < xhi; ++x)
                    acc = fmaxf(acc, trow[x]);
            }
            // out[r, c0+k, ph, pw]; lanes 0..48 -> 49 consecutive floats.
            out[(((size_t)r * C + (c0 + k)) * P_BINS + ph) * P_BINS + pw] = acc;
        }
        __syncthreads();   // all reads done before this buffer is re-filled
    }
}

extern "C" void kernel_launch(void* const* d_in, const int* in_sizes, int n_in,
                              void* d_out, int out_size, void* d_ws, size_t ws_size,
                              hipStream_t stream) {
    const float* feat = (const float*)d_in[0];
    const int*   rois = (const int*)d_in[1];
    float*       out  = (float*)d_out;

    const int C = in_sizes[0] / (H_FEAT * W_FEAT);   // 256
    const int R = in_sizes[1] / 4;                   // 256

    dim3 grid((unsigned)((C + CPB - 1) / CPB), (unsigned)R, 1);
    dim3 block(64, 1, 1);
    roi_maxpool_kernel<<<grid, block, 0, stream>>>(feat, rois, out, C, R);
}